// GraphAutoencoder_62706522521944
// MI455X (gfx1250) — compile-verified
//
#include <hip/hip_runtime.h>
#include <math.h>

// ---------------------------------------------------------------------------
// GCN encode: 3x (GCNConv -> act) over a fixed graph.
//   L1: 16->4 (relu), L2: 4->2 (relu), L3: 2->1 (tanh)
// deg/dinv computed once (graph is fixed across layers).
// Layer-1 x@W1 uses V_WMMA_F32_16X16X4_F32 (wave32, 16-row tiles, K=16 via
// 4 accumulating WMMAs), branchless fragment loads, LDS-transposed stores.
// Scatters use GLOBAL_ATOMIC_ADD_F32 via inline asm (native L2 f32 atomics,
// no CAS loops, non-returning -> STOREcnt; s_endpgm wait-idle + kernel
// boundaries provide inter-stage ordering).
// ---------------------------------------------------------------------------

typedef __attribute__((ext_vector_type(2))) float v2f;
typedef __attribute__((ext_vector_type(8))) float v8f;

__device__ __forceinline__ void atomic_add_f32(float* p, float v) {
  // Non-returning f32 atomic add straight to the L2 atomic units.
  asm volatile("global_atomic_add_f32 %0, %1, off" :: "v"(p), "v"(v) : "memory");
}

__global__ void zero_kernel(float* __restrict__ p, int n) {
  int i = blockIdx.x * blockDim.x + threadIdx.x;
  int stride = gridDim.x * blockDim.x;
  for (; i < n; i += stride) p[i] = 0.0f;
}

__global__ void degree_kernel(const int* __restrict__ col, float* __restrict__ deg, int E) {
  int e = blockIdx.x * blockDim.x + threadIdx.x;
  int stride = gridDim.x * blockDim.x;
  for (; e < E; e += stride) atomic_add_f32(&deg[col[e]], 1.0f);
}

__global__ void dinv_kernel(const float* __restrict__ deg, float* __restrict__ dinv,
                            float* __restrict__ invdeg, int N) {
  int i = blockIdx.x * blockDim.x + threadIdx.x;
  int stride = gridDim.x * blockDim.x;
  for (; i < N; i += stride) {
    float d = deg[i] + 1.0f;   // +1 self-loop
    dinv[i]   = rsqrtf(d);
    invdeg[i] = 1.0f / d;
  }
}

// xw1[N,4] = x[N,16] @ W1[16,4] using V_WMMA_F32_16X16X4_F32.
// One wave per 16-row tile. A(16x4 f32): lanes 0-15 hold K=k0,k0+1; lanes
// 16-31 hold K=k0+2,k0+3. B(4x16 f32): lane n holds W1[k][n&3] * (n<4).
// D(16x16 f32): vgpr v -> row v (lanes 0-15) / v+8 (lanes 16-31), col = lane&15.
// D tile is transposed through per-wave LDS into coalesced b128 row stores.
__global__ void xw1_wmma_kernel(const float* __restrict__ x, const float* __restrict__ W1,
                                float* __restrict__ xw1, int N) {
  __shared__ float lds[8 * 16 * 4];      // 8 waves/block * 16 rows * 4 cols

  const int lane = threadIdx.x & 31;
  const int wave = threadIdx.x >> 5;
  const int tile = blockIdx.x * (blockDim.x >> 5) + wave;
  const int numTiles = (N + 15) >> 4;
  if (tile >= numTiles) return;          // whole wave exits together (EXEC all-1 at WMMA)

  const int row_base = tile << 4;
  const int m   = lane & 15;
  const int khi = lane >> 4;             // 0: K pair {0,1}; 1: K pair {2,3} of each chunk
  // Clamp instead of predicate: out-of-range rows only feed D rows we never store.
  const int rowc = min(row_base + m, N - 1);

  const v2f* __restrict__ xv = (const v2f*)x;      // x row = 8 v2f pairs
  const float bm = (m < 4) ? 1.0f : 0.0f;          // B column mask (branchless)
  const int   bn = m & 3;

  v8f c = {0.f, 0.f, 0.f, 0.f, 0.f, 0.f, 0.f, 0.f};
#pragma unroll
  for (int kc = 0; kc < 4; ++kc) {
    const int k0 = (kc << 2) + (khi << 1);
    v2f a = xv[rowc * 8 + (kc << 1) + khi];        // unconditional global_load_b64
    v2f b;
    b.x = W1[k0 * 4 + bn] * bm;                    // unconditional dword loads, masked
    b.y = W1[(k0 + 1) * 4 + bn] * bm;
    c = __builtin_amdgcn_wmma_f32_16x16x4_f32(
        /*neg_a=*/false, a, /*neg_b=*/false, b,
        /*c_mod=*/(short)0, c, /*reuse_a=*/false, /*reuse_b=*/false);
  }

  // Transpose D through LDS (per-wave region; DS ops are in-order per wave).
  float* wlds = &lds[wave * 64];
  if (m < 4) {
    const int rhi = khi << 3;                      // +8 rows for lanes 16-31
#pragma unroll
    for (int v = 0; v < 8; ++v) wlds[(rhi + v) * 4 + m] = c[v];
  }
  if (lane < 16) {
    const int r = row_base + lane;
    if (r < N) {
      const float4 o = ((const float4*)wlds)[lane];
      ((float4*)xw1)[r] = o;                       // coalesced global_store_b128
    }
  }
}

// ---- edge scatter kernels: agg[col] += xw[row] * dinv[row] * dinv[col] ----

__global__ void scatter4_kernel(const int* __restrict__ row, const int* __restrict__ col,
                                const float* __restrict__ dinv, const float4* __restrict__ xw,
                                float* __restrict__ agg, int E) {
  int e = blockIdx.x * blockDim.x + threadIdx.x;
  int stride = gridDim.x * blockDim.x;
  for (; e < E; e += stride) {
    const int r = row[e], c = col[e];
    const float nrm = dinv[r] * dinv[c];
    const float4 v = xw[r];                 // 16B gather, L2 resident
    atomic_add_f32(&agg[c * 4 + 0], v.x * nrm);
    atomic_add_f32(&agg[c * 4 + 1], v.y * nrm);
    atomic_add_f32(&agg[c * 4 + 2], v.z * nrm);
    atomic_add_f32(&agg[c * 4 + 3], v.w * nrm);
  }
}

__global__ void scatter2_kernel(const int* __restrict__ row, const int* __restrict__ col,
                                const float* __restrict__ dinv, const float2* __restrict__ xw,
                                float* __restrict__ agg, int E) {
  int e = blockIdx.x * blockDim.x + threadIdx.x;
  int stride = gridDim.x * blockDim.x;
  for (; e < E; e += stride) {
    const int r = row[e], c = col[e];
    const float nrm = dinv[r] * dinv[c];
    const float2 v = xw[r];
    atomic_add_f32(&agg[c * 2 + 0], v.x * nrm);
    atomic_add_f32(&agg[c * 2 + 1], v.y * nrm);
  }
}

__global__ void scatter1_kernel(const int* __restrict__ row, const int* __restrict__ col,
                                const float* __restrict__ dinv, const float* __restrict__ xw,
                                float* __restrict__ agg, int E) {
  int e = blockIdx.x * blockDim.x + threadIdx.x;
  int stride = gridDim.x * blockDim.x;
  for (; e < E; e += stride) {
    const int r = row[e], c = col[e];
    atomic_add_f32(&agg[c], xw[r] * dinv[r] * dinv[c]);
  }
}

// ---- per-node epilogues: self-loop + bias + activation + next tiny matmul ----

__global__ void node1_kernel(const float4* __restrict__ agg1, const float4* __restrict__ xw1,
                             const float* __restrict__ invdeg, const float* __restrict__ b1,
                             const float* __restrict__ W2, float2* __restrict__ xw2, int N) {
  int i = blockIdx.x * blockDim.x + threadIdx.x;
  int stride = gridDim.x * blockDim.x;
  for (; i < N; i += stride) {
    const float4 a = agg1[i];
    const float4 w = xw1[i];
    const float id = invdeg[i];
    const float h0 = fmaxf(a.x + w.x * id + b1[0], 0.0f);
    const float h1 = fmaxf(a.y + w.y * id + b1[1], 0.0f);
    const float h2 = fmaxf(a.z + w.z * id + b1[2], 0.0f);
    const float h3 = fmaxf(a.w + w.w * id + b1[3], 0.0f);
    // xw2 = h @ W2, W2 is [4,2] row-major
    float2 o;
    o.x = h0 * W2[0] + h1 * W2[2] + h2 * W2[4] + h3 * W2[6];
    o.y = h0 * W2[1] + h1 * W2[3] + h2 * W2[5] + h3 * W2[7];
    xw2[i] = o;
  }
}

__global__ void node2_kernel(const float2* __restrict__ agg2, const float2* __restrict__ xw2,
                             const float* __restrict__ invdeg, const float* __restrict__ b2,
                             const float* __restrict__ W3, float* __restrict__ xw3, int N) {
  int i = blockIdx.x * blockDim.x + threadIdx.x;
  int stride = gridDim.x * blockDim.x;
  for (; i < N; i += stride) {
    const float2 a = agg2[i];
    const float2 w = xw2[i];
    const float id = invdeg[i];
    const float h0 = fmaxf(a.x + w.x * id + b2[0], 0.0f);
    const float h1 = fmaxf(a.y + w.y * id + b2[1], 0.0f);
    xw3[i] = h0 * W3[0] + h1 * W3[1];     // W3 is [2,1]
  }
}

__global__ void node3_kernel(const float* __restrict__ agg3, const float* __restrict__ xw3,
                             const float* __restrict__ invdeg, const float* __restrict__ b3,
                             float* __restrict__ out, int N) {
  int i = blockIdx.x * blockDim.x + threadIdx.x;
  int stride = gridDim.x * blockDim.x;
  for (; i < N; i += stride) {
    out[i] = tanhf(agg3[i] + xw3[i] * invdeg[i] + b3[0]);
  }
}

// ---------------------------------------------------------------------------

extern "C" void kernel_launch(void* const* d_in, const int* in_sizes, int n_in,
                              void* d_out, int out_size, void* d_ws, size_t ws_size,
                              hipStream_t stream) {
  const float* x  = (const float*)d_in[0];
  const int* edge = (const int*)d_in[1];
  const float* W1 = (const float*)d_in[2];
  const float* b1 = (const float*)d_in[3];
  const float* W2 = (const float*)d_in[4];
  const float* b2 = (const float*)d_in[5];
  const float* W3 = (const float*)d_in[6];
  const float* b3 = (const float*)d_in[7];

  const int N = in_sizes[0] / 16;  // 100000
  const int E = in_sizes[1] / 2;   // 3200000
  const int* row = edge;           // edge_index[0]
  const int* col = edge + E;       // edge_index[1]

  float* ws     = (float*)d_ws;    // layout (floats), N % 4 == 0 so 16B aligned:
  float* deg    = ws;              //  [0,   N)
  float* dinv   = ws + 1 * N;      //  [N,  2N)
  float* invdeg = ws + 2 * N;      //  [2N, 3N)
  float* xw1    = ws + 3 * N;      //  [3N, 7N)
  float* agg1   = ws + 7 * N;      //  [7N,11N)
  float* xw2    = ws + 11 * N;     //  [11N,13N)
  float* agg2   = ws + 13 * N;     //  [13N,15N)
  float* xw3    = ws + 15 * N;     //  [15N,16N)
  float* agg3   = ws + 16 * N;     //  [16N,17N)

  const int B = 256;               // 8 wave32s per block
  const int zn = 17 * N;
  const int gZ = (zn + B - 1) / B;
  const int gE = (E + B - 1) / B;
  const int gN = (N + B - 1) / B;

  zero_kernel<<<gZ, B, 0, stream>>>(ws, zn);
  degree_kernel<<<gE, B, 0, stream>>>(col, deg, E);
  dinv_kernel<<<gN, B, 0, stream>>>(deg, dinv, invdeg, N);

  // Layer 1: x @ W1 via WMMA, then scatter + epilogue(+ W2 matmul)
  const int tiles = (N + 15) / 16;
  const int wpb = B / 32;
  xw1_wmma_kernel<<<(tiles + wpb - 1) / wpb, B, 0, stream>>>(x, W1, xw1, N);
  scatter4_kernel<<<gE, B, 0, stream>>>(row, col, dinv, (const float4*)xw1, agg1, E);
  node1_kernel<<<gN, B, 0, stream>>>((const float4*)agg1, (const float4*)xw1, invdeg,
                                     b1, W2, (float2*)xw2, N);

  // Layer 2
  scatter2_kernel<<<gE, B, 0, stream>>>(row, col, dinv, (const float2*)xw2, agg2, E);
  node2_kernel<<<gN, B, 0, stream>>>((const float2*)agg2, (const float2*)xw2, invdeg,
                                     b2, W3, xw3, N);

  // Layer 3
  scatter1_kernel<<<gE, B, 0, stream>>>(row, col, dinv, xw3, agg3, E);
  node3_kernel<<<gN, B, 0, stream>>>(agg3, xw3, invdeg, b3, (float*)d_out, N);
}